// GNNProbeModel_79130477462155
// MI455X (gfx1250) — compile-verified
//
#include <hip/hip_runtime.h>
#include <hip/hip_bf16.h>

#define NH 128          // hidden dim == input dim
#define NLAYERS 3
#define LN_EPS 1e-5f

typedef float v2f __attribute__((ext_vector_type(2)));
typedef float v8f __attribute__((ext_vector_type(8)));

// ---------------------------------------------------------------------------
// guaranteed-native fp32 atomic add, device scope, no return (TH=0).
// gfx1250: GLOBAL_ATOMIC_ADD_F32, SCOPE_DEV so updates are coherent across WGPs.
// ---------------------------------------------------------------------------
__device__ __forceinline__ void atomic_add_f32_dev(float* p, float v) {
    asm volatile("global_atomic_add_f32 %0, %1, off scope:SCOPE_DEV"
                 :
                 : "v"(p), "v"(v)
                 : "memory");
}

// ---------------------------------------------------------------------------
// degree / norm precompute
// ---------------------------------------------------------------------------
__global__ void k_deg_init(float* deg, int n) {
    int i = blockIdx.x * 256 + threadIdx.x;
    if (i < n) deg[i] = 1.0f;                      // self-loop contribution
}

__global__ void k_deg_edges(const int* __restrict__ dst, float* deg, int e) {
    int i = blockIdx.x * 256 + threadIdx.x;
    if (i < e) atomic_add_f32_dev(&deg[dst[i]], 1.0f);
}

__global__ void k_dinv(float* deg_inout, int n) {
    int i = blockIdx.x * 256 + threadIdx.x;
    if (i < n) {
        float d = deg_inout[i];
        deg_inout[i] = (d > 0.0f) ? rsqrtf(d) : 0.0f;
    }
}

__global__ void k_norm(const int* __restrict__ src, const int* __restrict__ dst,
                       const float* __restrict__ dinv, float* __restrict__ nrm, int e) {
    int i = blockIdx.x * 256 + threadIdx.x;
    if (i < e) nrm[i] = dinv[src[i]] * dinv[dst[i]];
}

// ---------------------------------------------------------------------------
// fp32 WMMA GEMM: out[N,128] = A[N,128] @ W[128,128]
// block = 256 threads (8 waves), 16-row strip per block, 16 cols per wave.
// K-loop: 32 x v_wmma_f32_16x16x4_f32 accumulating in an 8-VGPR C fragment.
// ---------------------------------------------------------------------------
__global__ void __launch_bounds__(256)
k_gemm128(const float* __restrict__ A, const float* __restrict__ W,
          float* __restrict__ out) {
    __shared__ float lds[16 * 130];                // padded: kills bank conflicts
    const int row0 = blockIdx.x * 16;

    // cooperative stage of the 16x128 A tile (float2, coalesced)
    #pragma unroll
    for (int i = 0; i < 4; ++i) {
        int idx = threadIdx.x + 256 * i;           // 0..1023 -> 16 rows x 64 float2
        int r   = idx >> 6;
        int c2  = (idx & 63) * 2;
        const float2 v = *(const float2*)(A + (size_t)(row0 + r) * NH + c2);
        *(float2*)(lds + r * 130 + c2) = v;
    }
    __syncthreads();

    const int lane = threadIdx.x & 31;
    const int wv   = threadIdx.x >> 5;
    const int col0 = wv * 16;
    const int m    = lane & 15;                    // M (for A) / N (for B) index
    const int half = lane >> 4;                    // selects K pair

    const float* arow = lds + m * 130;
    v8f c;
    #pragma unroll
    for (int i = 0; i < 8; ++i) c[i] = 0.0f;

    #pragma unroll 4
    for (int kk = 0; kk < 32; ++kk) {
        const int k0 = kk * 4 + 2 * half;
        v2f a;                                     // A 16x4: lane m, K=k0..k0+1
        a.x = arow[k0];
        a.y = arow[k0 + 1];
        v2f b;                                     // B 4x16: col m, K=k0..k0+1
        b.x = W[(size_t)k0 * NH + col0 + m];
        b.y = W[(size_t)(k0 + 1) * NH + col0 + m];
        c = __builtin_amdgcn_wmma_f32_16x16x4_f32(false, a, false, b,
                                                  (short)0, c, false, false);
    }

    // C layout: lanes 0-15 -> M = r, lanes 16-31 -> M = 8 + r
    const int outRow = row0 + 8 * half;
    #pragma unroll
    for (int r = 0; r < 8; ++r)
        out[(size_t)(outRow + r) * NH + col0 + m] = c[r];
}

// ---------------------------------------------------------------------------
// agg[n][:] = b_conv[:] + hw[n][:] * dinv[n]^2       (bias + self-loop term)
// ---------------------------------------------------------------------------
__global__ void k_agg_init(const float* __restrict__ hw, const float* __restrict__ dinv,
                           const float* __restrict__ bias, float* __restrict__ agg, int n) {
    int t  = blockIdx.x * 256 + threadIdx.x;
    int nd = t >> 5;
    if (nd >= n) return;
    int c  = (t & 31) * 4;
    float di = dinv[nd];
    float sl = di * di;
    float4 v = *(const float4*)(hw + (size_t)nd * NH + c);
    float4 b = *(const float4*)(bias + c);
    float4 o;
    o.x = fmaf(v.x, sl, b.x); o.y = fmaf(v.y, sl, b.y);
    o.z = fmaf(v.z, sl, b.z); o.w = fmaf(v.w, sl, b.w);
    *(float4*)(agg + (size_t)nd * NH + c) = o;
}

// ---------------------------------------------------------------------------
// edge scatter: agg[dst] += hw[src] * norm        (wave per edge, float4/lane)
// ---------------------------------------------------------------------------
__global__ void k_scatter(const float* __restrict__ hw, const int* __restrict__ src,
                          const int* __restrict__ dst, const float* __restrict__ nrm,
                          float* __restrict__ agg, int e) {
    int t  = blockIdx.x * 256 + threadIdx.x;
    int ei = t >> 5;
    if (ei >= e) return;
    int c  = (t & 31) * 4;
    int s  = src[ei];
    int d  = dst[ei];
    float w = nrm[ei];
    float4 v = *(const float4*)(hw + (size_t)s * NH + c);
    float* p = agg + (size_t)d * NH + c;
    atomic_add_f32_dev(p + 0, v.x * w);
    atomic_add_f32_dev(p + 1, v.y * w);
    atomic_add_f32_dev(p + 2, v.z * w);
    atomic_add_f32_dev(p + 3, v.w * w);
}

// ---------------------------------------------------------------------------
// LayerNorm + exact GELU (+ optional bias, optional residual); wave per node
// ---------------------------------------------------------------------------
__device__ __forceinline__ float gelu_exact(float x) {
    return 0.5f * x * (1.0f + erff(x * 0.70710678118654752f));
}

__global__ void k_lngelu(const float* __restrict__ in, const float* __restrict__ bias,
                         const float* __restrict__ gamma, const float* __restrict__ beta,
                         const float* __restrict__ resid, float* __restrict__ out, int n) {
    const int lane = threadIdx.x & 31;
    const int wv   = threadIdx.x >> 5;
    const int node = blockIdx.x * 8 + wv;
    if (node >= n) return;
    const int c = lane * 4;

    float4 t = *(const float4*)(in + (size_t)node * NH + c);
    if (bias) {
        float4 b = *(const float4*)(bias + c);
        t.x += b.x; t.y += b.y; t.z += b.z; t.w += b.w;
    }
    float s  = t.x + t.y + t.z + t.w;
    float ss = fmaf(t.x, t.x, fmaf(t.y, t.y, fmaf(t.z, t.z, t.w * t.w)));
    #pragma unroll
    for (int o = 16; o > 0; o >>= 1) {
        s  += __shfl_xor(s,  o, 32);
        ss += __shfl_xor(ss, o, 32);
    }
    const float mu   = s * (1.0f / NH);
    const float var  = fmaf(ss, (1.0f / NH), -mu * mu);
    const float rstd = rsqrtf(var + LN_EPS);

    float4 g  = *(const float4*)(gamma + c);
    float4 be = *(const float4*)(beta + c);
    float4 o;
    o.x = gelu_exact(fmaf((t.x - mu) * rstd, g.x, be.x));
    o.y = gelu_exact(fmaf((t.y - mu) * rstd, g.y, be.y));
    o.z = gelu_exact(fmaf((t.z - mu) * rstd, g.z, be.z));
    o.w = gelu_exact(fmaf((t.w - mu) * rstd, g.w, be.w));
    if (resid) {
        float4 r = *(const float4*)(resid + (size_t)node * NH + c);
        o.x += r.x; o.y += r.y; o.z += r.z; o.w += r.w;
    }
    *(float4*)(out + (size_t)node * NH + c) = o;
}

// ---------------------------------------------------------------------------
// head: out[n] = dot(h[n], W_head) + b_head       (wave per node)
// ---------------------------------------------------------------------------
__global__ void k_head(const float* __restrict__ h, const float* __restrict__ wh,
                       const float* __restrict__ bh, float* __restrict__ out, int n) {
    const int lane = threadIdx.x & 31;
    const int wv   = threadIdx.x >> 5;
    const int node = blockIdx.x * 8 + wv;
    if (node >= n) return;
    const int c = lane * 4;
    float4 v = *(const float4*)(h + (size_t)node * NH + c);
    float4 w = *(const float4*)(wh + c);
    float p  = fmaf(v.x, w.x, fmaf(v.y, w.y, fmaf(v.z, w.z, v.w * w.w)));
    #pragma unroll
    for (int o = 16; o > 0; o >>= 1) p += __shfl_xor(p, o, 32);
    if (lane == 0) out[node] = p + bh[0];
}

// ---------------------------------------------------------------------------
extern "C" void kernel_launch(void* const* d_in, const int* in_sizes, int n_in,
                              void* d_out, int out_size, void* d_ws, size_t ws_size,
                              hipStream_t stream) {
    const float* x      = (const float*)d_in[0];
    const int*   ei     = (const int*)d_in[1];
    const float* W_in   = (const float*)d_in[2];
    const float* b_in   = (const float*)d_in[3];
    const float* g_in   = (const float*)d_in[4];
    const float* be_in  = (const float*)d_in[5];
    const float* W_conv = (const float*)d_in[6];
    const float* b_conv = (const float*)d_in[7];
    const float* g_ln   = (const float*)d_in[8];
    const float* be_ln  = (const float*)d_in[9];
    const float* W_head = (const float*)d_in[10];
    const float* b_head = (const float*)d_in[11];
    float* out = (float*)d_out;

    const int N = in_sizes[0] / NH;       // 100000
    const int E = in_sizes[1] / 2;        // 600000
    const int* src = ei;
    const int* dst = ei + E;

    char* w = (char*)d_ws;
    float* dinv = (float*)(w);                             // N floats
    float* nrm  = (float*)(w + (size_t)(1)   * (1 << 20)); // E floats
    float* h    = (float*)(w + (size_t)(4)   * (1 << 20)); // N*128 floats
    float* hw   = (float*)(w + (size_t)(60)  * (1 << 20)); // N*128 floats
    float* agg  = (float*)(w + (size_t)(116) * (1 << 20)); // N*128 floats

    const int gN   = (N + 255) / 256;
    const int gE   = (E + 255) / 256;
    const int gRow = N / 16;                  // 6250 (N divisible by 16)
    const int gNW  = (N * 32 + 255) / 256;    // wave-per-node kernels (float4/lane)
    const int gEW  = (E * 32 + 255) / 256;    // wave-per-edge scatter
    const int gNd  = (N + 7) / 8;             // 8 nodes per 256-thread block

    // degrees -> dinv -> per-edge norms (shared by all layers)
    k_deg_init<<<gN, 256, 0, stream>>>(dinv, N);
    k_deg_edges<<<gE, 256, 0, stream>>>(dst, dinv, E);
    k_dinv<<<gN, 256, 0, stream>>>(dinv, N);
    k_norm<<<gE, 256, 0, stream>>>(src, dst, dinv, nrm, E);

    // input projection + LN + GELU
    k_gemm128<<<gRow, 256, 0, stream>>>(x, W_in, hw);
    k_lngelu<<<gNd, 256, 0, stream>>>(hw, b_in, g_in, be_in, nullptr, h, N);

    // conv layers
    for (int i = 0; i < NLAYERS; ++i) {
        const float* Wc = W_conv + (size_t)i * NH * NH;
        const float* bc = b_conv + (size_t)i * NH;
        const float* gl = g_ln   + (size_t)i * NH;
        const float* bl = be_ln  + (size_t)i * NH;
        k_gemm128<<<gRow, 256, 0, stream>>>(h, Wc, hw);
        k_agg_init<<<gNW, 256, 0, stream>>>(hw, dinv, bc, agg, N);
        k_scatter<<<gEW, 256, 0, stream>>>(hw, src, dst, nrm, agg, E);
        k_lngelu<<<gNd, 256, 0, stream>>>(agg, nullptr, gl, bl, h, h, N);
    }

    // head
    k_head<<<gNd, 256, 0, stream>>>(h, W_head, b_head, out, N);
}